// ROIHeads_14293651161367
// MI455X (gfx1250) — compile-verified
//
#include <hip/hip_runtime.h>
#include <cstdint>

// ROI-heads label-and-sample for gfx1250 (MI455X).
// Kernel 1: per-proposal best-IoU match vs GT boxes staged in LDS via the
//           CDNA5 async global->LDS DMA path (ASYNCcnt).
// Kernel 2: per-image exact rank-based fg/bg sampling (one block per image),
//           priority row async-staged into LDS, rank loop register-blocked
//           with ds_load_b128 key fetches.

constexpr int NUM_CLASSES_K  = 80;
constexpr int BATCH_PER_IMG  = 512;
constexpr int NUM_FG_TARGET  = 128;
constexpr int CAND_BLK       = 8;   // candidates resident per thread per pass

__global__ void roi_match_kernel(const float* __restrict__ gt_boxes,   // [B,M,4]
                                 const float* __restrict__ prop_boxes, // [B,N,4]
                                 float* __restrict__ mvals,            // [B,K]
                                 int*   __restrict__ midxs,            // [B,K]
                                 int M, int N, int K) {
  const int img = blockIdx.y;
  extern __shared__ float4 gt_s[]; // M entries (M*16 bytes)

  const int tid = threadIdx.x;
  // ---- Stage GT boxes into LDS with CDNA5 async global->LDS DMA ----------
  if (tid < M) {
    const float* gsrc = gt_boxes + (size_t)(img * M + tid) * 4;
    // Low 32 bits of a generic pointer to LDS == LDS byte offset.
    uint32_t lds_off = (uint32_t)(uintptr_t)(&gt_s[tid]);
    asm volatile("global_load_async_to_lds_b128 %0, %1, off"
                 :: "v"(lds_off), "v"(gsrc)
                 : "memory");
  }
  asm volatile("s_wait_asynccnt 0x0" ::: "memory");
  __syncthreads();

  const int k = blockIdx.x * blockDim.x + tid;
  if (k >= K) return;

  float4 b;
  if (k < N) {
    b = *reinterpret_cast<const float4*>(prop_boxes + (size_t)(img * N + k) * 4);
  } else {
    b = gt_s[k - N]; // appended GT boxes
  }
  const float areaB = (b.z - b.x) * (b.w - b.y);

  float best = -1.0f;
  int   besti = 0;
  for (int m = 0; m < M; ++m) {
    float4 a = gt_s[m]; // broadcast read: all lanes hit the same LDS address
    float iw = fminf(a.z, b.z) - fmaxf(a.x, b.x);
    float ih = fminf(a.w, b.w) - fmaxf(a.y, b.y);
    iw = fmaxf(iw, 0.0f);
    ih = fmaxf(ih, 0.0f);
    float inter = iw * ih;
    float areaA = (a.z - a.x) * (a.w - a.y);
    float uni   = areaA + areaB - inter;
    float iou   = (inter > 0.0f) ? (inter / uni) : 0.0f;
    if (iou > best) { best = iou; besti = m; }  // first-occurrence argmax
  }
  mvals[(size_t)img * K + k] = best;   // best >= 0 after m=0
  midxs[(size_t)img * K + k] = besti;
}

__global__ __launch_bounds__(1024)
void roi_sample_kernel(const float* __restrict__ mvals,  // [B,K]
                       const int*   __restrict__ midxs,  // [B,K]
                       const float* __restrict__ pri,    // [B,K]
                       const int*   __restrict__ gt_cls, // [B,M]
                       float* __restrict__ out,          // 5 x [B,512]
                       int B, int M, int K) {
  const int img = blockIdx.x;
  const int tid = threadIdx.x;
  const int bs  = blockDim.x;
  extern __shared__ uint32_t key_s[]; // K words; holds pri bits, then keys
  __shared__ int fg_cnt_s;

  if (tid == 0) fg_cnt_s = 0;

  // ---- Async-stage this image's priority row (raw bits) into LDS ---------
  const float* prow = pri + (size_t)img * K;
  const int KV = K >> 2; // full 16B chunks
  for (int c = tid; c < KV; c += bs) {
    uint32_t lds_off = (uint32_t)(uintptr_t)(&key_s[c << 2]);
    const float* src = prow + (c << 2);
    asm volatile("global_load_async_to_lds_b128 %0, %1, off"
                 :: "v"(lds_off), "v"(src)
                 : "memory");
  }
  for (int t = (KV << 2) + tid; t < K; t += bs) {
    uint32_t lds_off = (uint32_t)(uintptr_t)(&key_s[t]);
    const float* src = prow + t;
    asm volatile("global_load_async_to_lds_b32 %0, %1, off"
                 :: "v"(lds_off), "v"(src)
                 : "memory");
  }
  asm volatile("s_wait_asynccnt 0x0" ::: "memory");
  __syncthreads();

  // ---- In-place transform: pri bits -> order-exact group key; count fg ----
  // pri in [0,1): float bits < 2^30, so key orders exactly like pri within a
  // group, and every fg key > every bg key.
  int local_fg = 0;
  for (int k = tid; k < K; k += bs) {
    float mv = mvals[(size_t)img * K + k];
    uint32_t key = key_s[k]; // raw float bits of pri[k] (own element: no race)
    if (mv >= 0.5f) { key |= (1u << 30); ++local_fg; }
    key_s[k] = key;
  }
  if (local_fg) atomicAdd(&fg_cnt_s, local_fg);
  __syncthreads();

  const int nfg_tot = fg_cnt_s;
  const int nfg = min(NUM_FG_TARGET, nfg_tot);
  const int nbg = min(BATCH_PER_IMG - nfg, K - nfg_tot);

  const size_t S = (size_t)B * BATCH_PER_IMG;
  float* out_iou = out;
  float* out_idx = out + S;
  float* out_cls = out + 2 * S;
  float* out_gt  = out + 3 * S;
  float* out_val = out + 4 * S;

  // ---- Defaults for invalid slots ----------------------------------------
  for (int s = tid; s < BATCH_PER_IMG; s += bs) {
    size_t o = (size_t)img * BATCH_PER_IMG + s;
    out_iou[o] = 0.0f;
    out_idx[o] = -1.0f;
    out_cls[o] = -1.0f;
    out_gt[o]  = -1.0f;
    out_val[o] = 0.0f;
  }
  __syncthreads();

  // ---- Exact descending rank within group (reversed-stable-argsort ties):
  //      rank_k = #{j: key_j > key_k || (key_j == key_k && j > k)}.
  //      Register-block CAND_BLK candidates; fetch keys 4-at-a-time (b128).
  const int K4 = K & ~3;
  for (int base = 0; base < K; base += CAND_BLK * bs) {
    uint32_t myk[CAND_BLK];
    int      kc[CAND_BLK];
    int      rk[CAND_BLK];
#pragma unroll
    for (int c = 0; c < CAND_BLK; ++c) {
      int k = base + tid + c * bs;
      kc[c]  = k;
      myk[c] = (k < K) ? key_s[k] : 0xFFFFFFFFu; // unreachable key -> rank 0
      rk[c]  = 0;
    }
    for (int j = 0; j < K4; j += 4) {
      uint4 kv = *reinterpret_cast<const uint4*>(&key_s[j]); // ds_load_b128
#pragma unroll
      for (int c = 0; c < CAND_BLK; ++c) {
        rk[c] += (kv.x > myk[c]) || ((kv.x == myk[c]) && ((j + 0) > kc[c]));
        rk[c] += (kv.y > myk[c]) || ((kv.y == myk[c]) && ((j + 1) > kc[c]));
        rk[c] += (kv.z > myk[c]) || ((kv.z == myk[c]) && ((j + 2) > kc[c]));
        rk[c] += (kv.w > myk[c]) || ((kv.w == myk[c]) && ((j + 3) > kc[c]));
      }
    }
    for (int j = K4; j < K; ++j) {
      uint32_t kj = key_s[j];
#pragma unroll
      for (int c = 0; c < CAND_BLK; ++c)
        rk[c] += (kj > myk[c]) || ((kj == myk[c]) && (j > kc[c]));
    }
#pragma unroll
    for (int c = 0; c < CAND_BLK; ++c) {
      int k = kc[c];
      if (k >= K) continue;
      const bool fg = (myk[c] >> 30) & 1u;
      int rank = rk[c];
      int slot = -1;
      if (fg) {
        if (rank < nfg) slot = rank;
      } else {
        rank -= nfg_tot; // all fg keys rank above any bg key
        if (rank < nbg) slot = nfg + rank;
      }
      if (slot >= 0) {
        size_t o = (size_t)img * BATCH_PER_IMG + slot;
        float mv = mvals[(size_t)img * K + k];
        int   mi = midxs[(size_t)img * K + k];
        out_iou[o] = mv;
        out_idx[o] = (float)k;
        out_cls[o] = fg ? (float)gt_cls[img * M + mi] : (float)NUM_CLASSES_K;
        out_gt[o]  = (float)mi;
        out_val[o] = 1.0f;
      }
    }
  }
}

extern "C" void kernel_launch(void* const* d_in, const int* in_sizes, int n_in,
                              void* d_out, int out_size, void* d_ws, size_t ws_size,
                              hipStream_t stream) {
  // Derive shapes from sizes (reference: B=64, M=100, N=4000, K=4100).
  const int B = out_size / (5 * BATCH_PER_IMG);
  const int M = in_sizes[2] / B;           // gt_classes:      B*M
  const int N = in_sizes[1] / (4 * B);     // proposal_boxes:  B*N*4
  const int K = in_sizes[3] / B;           // rand_priority:   B*K (== N+M)

  const float* gt_boxes   = (const float*)d_in[0];
  const float* prop_boxes = (const float*)d_in[1];
  const int*   gt_cls     = (const int*)d_in[2];
  const float* pri        = (const float*)d_in[3];
  float*       out        = (float*)d_out;

  // Workspace: matched_vals [B*K] floats, matched_idxs [B*K] ints (~2.1 MB).
  float* mv = (float*)d_ws;
  int*   mi = (int*)(mv + (size_t)B * K);

  dim3 gA((K + 255) / 256, B);
  roi_match_kernel<<<gA, 256, (size_t)M * sizeof(float4), stream>>>(
      gt_boxes, prop_boxes, mv, mi, M, N, K);

  roi_sample_kernel<<<B, 1024, (size_t)K * sizeof(uint32_t), stream>>>(
      mv, mi, pri, gt_cls, out, B, M, K);
}